// TransformerModel_28948079575191
// MI455X (gfx1250) — compile-verified
//
#include <hip/hip_runtime.h>

#define N_NODES 50000
#define N_EDGES 800000
#define IN_F    128
#define HID_F   64
#define HEADS   4
#define OUT_F   32
#define QKV_W   (HEADS * HID_F)   // 256

typedef __attribute__((ext_vector_type(2))) float v2f;
typedef __attribute__((ext_vector_type(8))) float v8f;

// ---- order-preserving float <-> uint for atomicMax-based segment max ----
__device__ __forceinline__ unsigned flt_flip(float f) {
  unsigned u = __float_as_uint(f);
  return (u & 0x80000000u) ? ~u : (u | 0x80000000u);
}
__device__ __forceinline__ float flt_unflip(unsigned u) {
  unsigned v = (u & 0x80000000u) ? (u & 0x7fffffffu) : ~u;
  return __uint_as_float(v);
}

// ============================================================================
// f32 WMMA GEMM:  C[M,Nw] = A[M,K] @ W[K,Nw] + bias[Nw]
// One wave computes one 16x16 tile using V_WMMA_F32_16X16X4_F32.
// A-frag layout (16x4 f32): lanes 0-15 hold K=0,1 ; lanes 16-31 hold K=2,3.
// B-frag layout (4x16 f32): mirrored (row striped across lanes per VGPR).
// C/D layout: VGPR r = row r (lanes 0-15) / row r+8 (lanes 16-31).
// All dims are multiples of 16 in this model -> no edge predication, EXEC=all-1.
// ============================================================================
__global__ void __launch_bounds__(128)
wmma_gemm_bias(const float* __restrict__ A, const float* __restrict__ W,
               const float* __restrict__ bias, float* __restrict__ C,
               int M, int K, int Nw) {
  const int tilesN = Nw >> 4;
  const int tilesM = M >> 4;
  const int wave   = blockIdx.x * (blockDim.x >> 5) + (threadIdx.x >> 5);
  if (wave >= tilesM * tilesN) return;            // wave-uniform exit
  const int tm   = wave / tilesN;
  const int tn   = wave % tilesN;
  const int lane = threadIdx.x & 31;
  const int half = lane >> 4;                     // K-half select
  const int l15  = lane & 15;

  const float* Arow = A + (size_t)(tm * 16 + l15) * K;
  const float* Bcol = W + (tn * 16 + l15);

  v8f c = {};
  #pragma unroll 4
  for (int k = 0; k < K; k += 4) {
    const int kk = k + half * 2;
    v2f a, b;
    a.x = Arow[kk];
    a.y = Arow[kk + 1];
    b.x = Bcol[(size_t)kk * Nw];
    b.y = Bcol[(size_t)(kk + 1) * Nw];
    c = __builtin_amdgcn_wmma_f32_16x16x4_f32(
        /*neg_a=*/false, a, /*neg_b=*/false, b,
        /*c_mod=*/(short)0, c, /*reuse_a=*/false, /*reuse_b=*/false);
  }

  const int col  = tn * 16 + l15;
  const int row0 = tm * 16 + half * 8;
  const float bv = bias[col];
  #pragma unroll
  for (int r = 0; r < 8; ++r)
    C[(size_t)(row0 + r) * Nw + col] = c[r] + bv;
}

// ============================================================================
// Edge phase kernels (segment softmax attention). One thread per (edge, head).
// ============================================================================
__global__ void __launch_bounds__(256)
edge_logits(const float* __restrict__ q, const float* __restrict__ k,
            const int* __restrict__ src, const int* __restrict__ dst,
            unsigned* __restrict__ mEnc, float* __restrict__ elog, int nE) {
  int t = blockIdx.x * blockDim.x + threadIdx.x;
  if (t >= nE * HEADS) return;
  const int e = t >> 2, h = t & 3;
  const int s = src[e], d = dst[e];
  const float4* qp = (const float4*)(q + (size_t)d * QKV_W + h * HID_F);
  const float4* kp = (const float4*)(k + (size_t)s * QKV_W + h * HID_F);
  float acc = 0.f;
  #pragma unroll
  for (int i = 0; i < HID_F / 4; ++i) {
    float4 a = qp[i], b = kp[i];
    acc += a.x * b.x + a.y * b.y + a.z * b.z + a.w * b.w;
  }
  acc *= 0.125f;                                  // 1/sqrt(64)
  elog[t] = acc;
  atomicMax(&mEnc[(size_t)d * HEADS + h], flt_flip(acc));
}

__global__ void __launch_bounds__(256)
edge_expsum(const int* __restrict__ dst, const unsigned* __restrict__ mEnc,
            float* __restrict__ elog, float* __restrict__ ssum, int nE) {
  int t = blockIdx.x * blockDim.x + threadIdx.x;
  if (t >= nE * HEADS) return;
  const int e = t >> 2, h = t & 3;
  const int d = dst[e];
  const float m  = flt_unflip(mEnc[(size_t)d * HEADS + h]);
  const float ex = __expf(elog[t] - m);
  elog[t] = ex;
  atomicAdd(&ssum[(size_t)d * HEADS + h], ex);
}

__global__ void __launch_bounds__(256)
edge_agg(const int* __restrict__ src, const int* __restrict__ dst,
         const float* __restrict__ v, const float* __restrict__ elog,
         const float* __restrict__ ssum, float* __restrict__ agg, int nE) {
  int t = blockIdx.x * blockDim.x + threadIdx.x;
  if (t >= nE * HEADS) return;
  const int e = t >> 2, h = t & 3;
  const int s = src[e], d = dst[e];
  const float alpha = elog[t] / ssum[(size_t)d * HEADS + h];
  const float4* vp = (const float4*)(v + (size_t)s * QKV_W + h * HID_F);
  float* ap = agg + (size_t)d * QKV_W + h * HID_F;
  #pragma unroll
  for (int i = 0; i < HID_F / 4; ++i) {
    float4 vv = vp[i];
    atomicAdd(ap + 4 * i + 0, alpha * vv.x);
    atomicAdd(ap + 4 * i + 1, alpha * vv.y);
    atomicAdd(ap + 4 * i + 2, alpha * vv.z);
    atomicAdd(ap + 4 * i + 3, alpha * vv.w);
  }
}

// head-mean + skip + LayerNorm (+optional ReLU); one thread per node
__global__ void __launch_bounds__(256)
node_update(const float* __restrict__ agg, const float* __restrict__ skip,
            const float* __restrict__ g, const float* __restrict__ b,
            float* __restrict__ out, int n, int do_relu) {
  int i = blockIdx.x * blockDim.x + threadIdx.x;
  if (i >= n) return;
  const float* a  = agg  + (size_t)i * QKV_W;
  const float* sk = skip + (size_t)i * HID_F;
  float sum = 0.f, sq = 0.f;
  for (int d = 0; d < HID_F; ++d) {
    float val = 0.25f * (a[d] + a[HID_F + d] + a[2 * HID_F + d] + a[3 * HID_F + d]) + sk[d];
    sum += val;
    sq  += val * val;
  }
  const float mu  = sum * (1.f / HID_F);
  const float var = sq * (1.f / HID_F) - mu * mu;
  const float inv = rsqrtf(var + 1e-5f);
  float* o = out + (size_t)i * HID_F;
  for (int d = 0; d < HID_F; ++d) {
    float val = 0.25f * (a[d] + a[HID_F + d] + a[2 * HID_F + d] + a[3 * HID_F + d]) + sk[d];
    float y = (val - mu) * inv * g[d] + b[d];
    o[d] = do_relu ? fmaxf(y, 0.f) : y;
  }
}

// ============================================================================
// Host launch
// ============================================================================
static inline void gemm(const float* A, const float* W, const float* bias,
                        float* C, int M, int K, int Nw, hipStream_t s) {
  int tiles  = (M >> 4) * (Nw >> 4);
  int blocks = (tiles + 3) / 4;                   // 4 waves / 128-thread block
  wmma_gemm_bias<<<blocks, 128, 0, s>>>(A, W, bias, C, M, K, Nw);
}

extern "C" void kernel_launch(void* const* d_in, const int* in_sizes, int n_in,
                              void* d_out, int out_size, void* d_ws, size_t ws_size,
                              hipStream_t stream) {
  const float* x   = (const float*)d_in[0];
  const int*   ei  = (const int*)d_in[1];
  const int*   src = ei;
  const int*   dst = ei + N_EDGES;
  const float *Wq1 = (const float*)d_in[2],  *bq1 = (const float*)d_in[3];
  const float *Wk1 = (const float*)d_in[4],  *bk1 = (const float*)d_in[5];
  const float *Wv1 = (const float*)d_in[6],  *bv1 = (const float*)d_in[7];
  const float *Ws1 = (const float*)d_in[8],  *bs1 = (const float*)d_in[9];
  const float *Wq2 = (const float*)d_in[10], *bq2 = (const float*)d_in[11];
  const float *Wk2 = (const float*)d_in[12], *bk2 = (const float*)d_in[13];
  const float *Wv2 = (const float*)d_in[14], *bv2 = (const float*)d_in[15];
  const float *Ws2 = (const float*)d_in[16], *bs2 = (const float*)d_in[17];
  const float *lng = (const float*)d_in[18], *lnb = (const float*)d_in[19];
  const float *Wf  = (const float*)d_in[20], *bf  = (const float*)d_in[21];

  // ---- carve workspace ----
  char* ws = (char*)d_ws;
  size_t off = 0;
  auto carve = [&](size_t bytes) -> void* {
    void* p = ws + off;
    off += (bytes + 255) & ~(size_t)255;
    return p;
  };
  float*    qb   = (float*)   carve((size_t)N_NODES * QKV_W * 4);
  float*    kb   = (float*)   carve((size_t)N_NODES * QKV_W * 4);
  float*    vb   = (float*)   carve((size_t)N_NODES * QKV_W * 4);
  float*    agg  = (float*)   carve((size_t)N_NODES * QKV_W * 4);
  float*    skb  = (float*)   carve((size_t)N_NODES * HID_F * 4);
  float*    h1   = (float*)   carve((size_t)N_NODES * HID_F * 4);
  float*    h2   = (float*)   carve((size_t)N_NODES * HID_F * 4);
  float*    elog = (float*)   carve((size_t)N_EDGES * HEADS * 4);
  unsigned* mEnc = (unsigned*)carve((size_t)N_NODES * HEADS * 4);
  float*    ssum = (float*)   carve((size_t)N_NODES * HEADS * 4);
  (void)ws_size; (void)in_sizes; (void)n_in; (void)out_size;

  const int egrid = (N_EDGES * HEADS + 255) / 256;
  const int ngrid = (N_NODES + 255) / 256;

  // -------- layer 1 --------
  gemm(x, Wq1, bq1, qb,  N_NODES, IN_F, QKV_W, stream);
  gemm(x, Wk1, bk1, kb,  N_NODES, IN_F, QKV_W, stream);
  gemm(x, Wv1, bv1, vb,  N_NODES, IN_F, QKV_W, stream);
  gemm(x, Ws1, bs1, skb, N_NODES, IN_F, HID_F, stream);

  hipMemsetAsync(mEnc, 0, (size_t)N_NODES * HEADS * 4, stream);
  hipMemsetAsync(ssum, 0, (size_t)N_NODES * HEADS * 4, stream);
  hipMemsetAsync(agg,  0, (size_t)N_NODES * QKV_W * 4, stream);

  edge_logits<<<egrid, 256, 0, stream>>>(qb, kb, src, dst, mEnc, elog, N_EDGES);
  edge_expsum<<<egrid, 256, 0, stream>>>(dst, mEnc, elog, ssum, N_EDGES);
  edge_agg   <<<egrid, 256, 0, stream>>>(src, dst, vb, elog, ssum, agg, N_EDGES);
  node_update<<<ngrid, 256, 0, stream>>>(agg, skb, lng, lnb, h1, N_NODES, 1);

  // -------- layer 2 --------
  gemm(h1, Wq2, bq2, qb,  N_NODES, HID_F, QKV_W, stream);
  gemm(h1, Wk2, bk2, kb,  N_NODES, HID_F, QKV_W, stream);
  gemm(h1, Wv2, bv2, vb,  N_NODES, HID_F, QKV_W, stream);
  gemm(h1, Ws2, bs2, skb, N_NODES, HID_F, HID_F, stream);

  hipMemsetAsync(mEnc, 0, (size_t)N_NODES * HEADS * 4, stream);
  hipMemsetAsync(ssum, 0, (size_t)N_NODES * HEADS * 4, stream);
  hipMemsetAsync(agg,  0, (size_t)N_NODES * QKV_W * 4, stream);

  edge_logits<<<egrid, 256, 0, stream>>>(qb, kb, src, dst, mEnc, elog, N_EDGES);
  edge_expsum<<<egrid, 256, 0, stream>>>(dst, mEnc, elog, ssum, N_EDGES);
  edge_agg   <<<egrid, 256, 0, stream>>>(src, dst, vb, elog, ssum, agg, N_EDGES);
  node_update<<<ngrid, 256, 0, stream>>>(agg, skb, lng, lnb, h2, N_NODES, 0);

  // -------- final projection --------
  gemm(h2, Wf, bf, (float*)d_out, N_NODES, HID_F, OUT_F, stream);
}